// SelfAttentionScores_20736102105577
// MI455X (gfx1250) — compile-verified
//
#include <hip/hip_runtime.h>
#include <math.h>

typedef __attribute__((ext_vector_type(2))) float v2f;
typedef __attribute__((ext_vector_type(4))) float v4f;
typedef __attribute__((ext_vector_type(8))) float v8f;

#define SEQ       2048
#define HDIM      64
#define QROWS_WG  128     // query rows per workgroup (8 waves x 16)
#define KCHUNK    128     // keys staged in LDS per iteration
#define LDSTRIDE  68      // 64 + 4 pad floats -> conflict-free b64 reads
#define NKT       (KCHUNK / 16)

__launch_bounds__(256)
__global__ void attn_scores_wmma_f32(const float* __restrict__ qg,
                                     const float* __restrict__ kg,
                                     float* __restrict__ outg) {
    __shared__ float ksh[KCHUNK * LDSTRIDE];   // 34,816 B

    const float scale = 0.125f;                // 1/sqrt(64)

    const int tid  = threadIdx.x;
    const int lane = tid & 31;
    const int wave = tid >> 5;
    const int l16  = lane & 15;
    const int half = lane >> 4;                // 0: K={0,1}, 1: K={2,3}
    const int koff = half * 2;

    const int qglobal = blockIdx.x * QROWS_WG; // flat query row over (b,h,s)
    const int bh      = qglobal >> 11;         // SEQ == 2048
    const int q0      = qglobal & (SEQ - 1);

    const float* __restrict__ Qb = qg  + (size_t)bh * SEQ * HDIM;
    const float* __restrict__ Kb = kg  + (size_t)bh * SEQ * HDIM;
    float*       __restrict__ Ob = outg + (size_t)bh * SEQ * SEQ;

    // ---- Load this wave's 16x64 Q tile into registers (A-operand layout) ----
    // A 16x4 f32: lanes 0-15 rows M=0..15 hold K=0,1 ; lanes 16-31 hold K=2,3
    const int qrow = q0 + wave * 16 + l16;
    v2f qa[16];
#pragma unroll
    for (int c = 0; c < 16; ++c)
        qa[c] = *(const v2f*)(Qb + qrow * HDIM + 4 * c + koff);

    // Per-lane online-softmax state for the 8 rows this lane touches
    float mrow[8], srow[8];
#pragma unroll
    for (int r = 0; r < 8; ++r) { mrow[r] = -INFINITY; srow[r] = 0.0f; }

    // ================= PASS 1: online row max / sum =================
    for (int chunk = 0; chunk < SEQ; chunk += KCHUNK) {
        // stage K chunk -> LDS (coalesced b128 loads)
#pragma unroll
        for (int i = 0; i < (KCHUNK * HDIM) / (256 * 4); ++i) {
            int idx = i * 1024 + tid * 4;
            v4f v = *(const v4f*)(Kb + chunk * HDIM + idx);
            *(v4f*)(&ksh[(idx >> 6) * LDSTRIDE + (idx & 63)]) = v;
        }
        // prefetch next chunk while we compute
        if (chunk + KCHUNK < SEQ)
            __builtin_prefetch(Kb + (chunk + KCHUNK) * HDIM + tid * 4, 0, 0);
        __syncthreads();

#pragma unroll
        for (int kt = 0; kt < NKT; ++kt) {
            const float* kr = &ksh[(kt * 16 + l16) * LDSTRIDE + koff];
            v8f acc = {};
#pragma unroll
            for (int c = 0; c < 16; ++c) {
                v2f b; b.x = kr[4 * c]; b.y = kr[4 * c + 1];
                acc = __builtin_amdgcn_wmma_f32_16x16x4_f32(
                        false, qa[c], false, b, (short)0, acc, false, false);
            }
#pragma unroll
            for (int r = 0; r < 8; ++r) {
                float x  = acc[r] * scale;
                float mo = mrow[r];
                float mn = fmaxf(mo, x);
                srow[r]  = srow[r] * __expf(mo - mn) + __expf(x - mn);
                mrow[r]  = mn;
            }
        }
        __syncthreads();
    }

    // ---- combine (m,s) across the 16 lanes sharing each row ----
#pragma unroll
    for (int r = 0; r < 8; ++r) {
#pragma unroll
        for (int off = 1; off < 16; off <<= 1) {
            float mo = __shfl_xor(mrow[r], off, 32);
            float so = __shfl_xor(srow[r], off, 32);
            float mn = fmaxf(mrow[r], mo);
            srow[r]  = srow[r] * __expf(mrow[r] - mn) + so * __expf(mo - mn);
            mrow[r]  = mn;
        }
        srow[r] = 1.0f / srow[r];   // now holds 1/rowsum
    }

    // ================= PASS 2: recompute, normalize, write =================
    // C/D layout: VGPR r -> row M = r + 8*half, col N = l16
    float* orow0 = Ob + (size_t)(q0 + wave * 16 + 8 * half) * SEQ + l16;

    for (int chunk = 0; chunk < SEQ; chunk += KCHUNK) {
#pragma unroll
        for (int i = 0; i < (KCHUNK * HDIM) / (256 * 4); ++i) {
            int idx = i * 1024 + tid * 4;
            v4f v = *(const v4f*)(Kb + chunk * HDIM + idx);
            *(v4f*)(&ksh[(idx >> 6) * LDSTRIDE + (idx & 63)]) = v;
        }
        __syncthreads();

#pragma unroll
        for (int kt = 0; kt < NKT; ++kt) {
            const float* kr = &ksh[(kt * 16 + l16) * LDSTRIDE + koff];
            v8f acc = {};
#pragma unroll
            for (int c = 0; c < 16; ++c) {
                v2f b; b.x = kr[4 * c]; b.y = kr[4 * c + 1];
                acc = __builtin_amdgcn_wmma_f32_16x16x4_f32(
                        false, qa[c], false, b, (short)0, acc, false, false);
            }
            float* op = orow0 + chunk + kt * 16;
#pragma unroll
            for (int r = 0; r < 8; ++r) {
                float x = acc[r] * scale;
                op[(size_t)r * SEQ] = __expf(x - mrow[r]) * srow[r];
            }
        }
        __syncthreads();
    }
}

extern "C" void kernel_launch(void* const* d_in, const int* in_sizes, int n_in,
                              void* d_out, int out_size, void* d_ws, size_t ws_size,
                              hipStream_t stream) {
    const float* q = (const float*)d_in[0];
    const float* k = (const float*)d_in[1];
    float* out = (float*)d_out;
    (void)in_sizes; (void)n_in; (void)out_size; (void)d_ws; (void)ws_size;

    const int total_qrows = 4 * 16 * SEQ;            // B*H*S
    dim3 grid(total_qrows / QROWS_WG);               // 1024
    dim3 block(256);
    attn_scores_wmma_f32<<<grid, block, 0, stream>>>(q, k, out);
}